// CantorMultiheadFusion_34875134444337
// MI455X (gfx1250) — compile-verified
//
#include <hip/hip_runtime.h>
#include <hip/hip_bf16.h>

typedef __attribute__((ext_vector_type(2))) float v2f;
typedef __attribute__((ext_vector_type(8))) float v8f;

#define S_TOK 2048
#define DIM   1024
#define NHEAD 16
#define KNBR  64
#define HDIM  64

#define WMMA_F32(a, b, c) \
    __builtin_amdgcn_wmma_f32_16x16x4_f32(false, (a), false, (b), (short)0, (c), false, false)

// ---------------------------------------------------------------------------
// C[M,N] = A[M,K] @ B[N,K]^T   (+ bias[N] + resid[M,N] when non-null)
// One wave computes a 32(M) x 32(N) tile (2x2 grid of 16x16 WMMA subtiles).
//
// Distance-2 software pipeline: chunk k+8 is loaded while chunk k computes.
// __builtin_amdgcn_sched_barrier(0) pins the prefetch loads ABOVE the WMMA
// block so the scheduler cannot sink them back to their uses; the waitcnt
// pass then leaves the prefetches outstanding across the compute stage.
//
// fp32 WMMA operand layouts (ISA 7.12.2, wave32):
//   A 16x4 : lane l -> M = l&15, VGPR v holds K = 2*(l>>4) + v
//   B 4x16 : lane l -> N = l&15, VGPR v holds K = 2*(l>>4) + v
//   C/D    : lane l -> N = l&15, VGPR v holds M = v + 8*(l>>4)
// ---------------------------------------------------------------------------
__global__ __launch_bounds__(128)
void gemm_nt_wmma(const float* __restrict__ A, const float* __restrict__ Bt,
                  float* __restrict__ C, int M, int N, int Kd,
                  const float* __restrict__ bias, const float* __restrict__ resid)
{
    const int lane  = threadIdx.x & 31;
    const int wave  = threadIdx.x >> 5;
    const int gwave = blockIdx.x * 4 + wave;
    const int tilesN = N >> 5;                 // 32-wide N tiles
    const int tm = gwave / tilesN;
    const int tn = gwave % tilesN;
    if (tm * 32 >= M) return;

    const int half = lane >> 4;                // 0 or 1
    const int l16  = lane & 15;
    const int koff = 2 * half;

    const float* pa0 = A  + (size_t)(tm * 32 + l16) * Kd + koff;
    const float* pa1 = pa0 + (size_t)16 * Kd;
    const float* pb0 = Bt + (size_t)(tn * 32 + l16) * Kd + koff;
    const float* pb1 = pb0 + (size_t)16 * Kd;

    v8f acc00 = {}; v8f acc01 = {}; v8f acc10 = {}; v8f acc11 = {};

    // pipeline prologue: chunk 0 -> cur, chunk 1 -> nxt
    v2f a0c = *(const v2f*)(pa0);
    v2f a1c = *(const v2f*)(pa1);
    v2f b0c = *(const v2f*)(pb0);
    v2f b1c = *(const v2f*)(pb1);
    v2f a0n = *(const v2f*)(pa0 + 4);
    v2f a1n = *(const v2f*)(pa1 + 4);
    v2f b0n = *(const v2f*)(pb0 + 4);
    v2f b1n = *(const v2f*)(pb1 + 4);

    #pragma unroll 4
    for (int kk = 8; kk < Kd; kk += 4) {
        // prefetch chunk kk (2 stages ahead of the compute below)
        v2f a0f = *(const v2f*)(pa0 + kk);
        v2f a1f = *(const v2f*)(pa1 + kk);
        v2f b0f = *(const v2f*)(pb0 + kk);
        v2f b1f = *(const v2f*)(pb1 + kk);

        __builtin_amdgcn_sched_barrier(0);   // keep prefetches above WMMAs

        acc00 = WMMA_F32(a0c, b0c, acc00);
        acc01 = WMMA_F32(a0c, b1c, acc01);
        acc10 = WMMA_F32(a1c, b0c, acc10);
        acc11 = WMMA_F32(a1c, b1c, acc11);

        a0c = a0n; a1c = a1n; b0c = b0n; b1c = b1n;
        a0n = a0f; a1n = a1f; b0n = b0f; b1n = b1f;
    }

    // drain the two in-flight stages
    acc00 = WMMA_F32(a0c, b0c, acc00);
    acc01 = WMMA_F32(a0c, b1c, acc01);
    acc10 = WMMA_F32(a1c, b0c, acc10);
    acc11 = WMMA_F32(a1c, b1c, acc11);

    acc00 = WMMA_F32(a0n, b0n, acc00);
    acc01 = WMMA_F32(a0n, b1n, acc01);
    acc10 = WMMA_F32(a1n, b0n, acc10);
    acc11 = WMMA_F32(a1n, b1n, acc11);

    const int nb0 = tn * 32 + l16;
    const int nb1 = nb0 + 16;
    #pragma unroll
    for (int v = 0; v < 8; ++v) {
        const int m0 = tm * 32 + v + 8 * half;       // rows of acc0x
        const int m1 = m0 + 16;                      // rows of acc1x
        const size_t r0 = (size_t)m0 * N;
        const size_t r1 = (size_t)m1 * N;
        float c00 = acc00[v], c01 = acc01[v], c10 = acc10[v], c11 = acc11[v];
        if (bias) {
            c00 += bias[nb0]; c01 += bias[nb1];
            c10 += bias[nb0]; c11 += bias[nb1];
        }
        if (resid) {
            c00 += resid[r0 + nb0]; c01 += resid[r0 + nb1];
            c10 += resid[r1 + nb0]; c11 += resid[r1 + nb1];
        }
        C[r0 + nb0] = c00;
        C[r0 + nb1] = c01;
        C[r1 + nb0] = c10;
        C[r1 + nb1] = c11;
    }
}

// ---------------------------------------------------------------------------
// Gather-attention: one block per token, one wave (32 lanes) per head.
// Phase 1: lane owns neighbors k and k+32 -> scores via float4 dot products.
// Softmax via wave32 shuffle reductions. Phase 2: lane owns dims 2*lane,
// 2*lane+1; loop over all 64 neighbors broadcasting w_k with __shfl
// (coalesced float2 neighbor loads per k).
// ---------------------------------------------------------------------------
__global__ __launch_bounds__(512)
void attn_gather(const float* __restrict__ h, const int* __restrict__ routes,
                 float* __restrict__ fused)
{
    const int s    = blockIdx.x;
    const int head = threadIdx.x >> 5;
    const int lane = threadIdx.x & 31;

    const float* q = h + (size_t)s * DIM + head * HDIM;
    const int*   r = routes + (size_t)s * KNBR;

    const int k0 = lane;
    const int k1 = lane + 32;
    const float* n0 = h + (size_t)r[k0] * DIM + head * HDIM;
    const float* n1 = h + (size_t)r[k1] * DIM + head * HDIM;

    float s0 = 0.f, s1 = 0.f;
    #pragma unroll
    for (int d = 0; d < HDIM; d += 4) {
        float4 q4 = *(const float4*)(q  + d);
        float4 a4 = *(const float4*)(n0 + d);
        float4 b4 = *(const float4*)(n1 + d);
        s0 += q4.x * a4.x + q4.y * a4.y + q4.z * a4.z + q4.w * a4.w;
        s1 += q4.x * b4.x + q4.y * b4.y + q4.z * b4.z + q4.w * b4.w;
    }
    s0 *= 0.125f;   // 1/sqrt(64)
    s1 *= 0.125f;

    // wave32 max reduction
    float mx = fmaxf(s0, s1);
    #pragma unroll
    for (int off = 16; off > 0; off >>= 1)
        mx = fmaxf(mx, __shfl_xor(mx, off, 32));

    float e0 = __expf(s0 - mx);
    float e1 = __expf(s1 - mx);
    float sum = e0 + e1;
    #pragma unroll
    for (int off = 16; off > 0; off >>= 1)
        sum += __shfl_xor(sum, off, 32);

    const float inv = 1.f / sum;
    const float w0 = e0 * inv;
    const float w1 = e1 * inv;

    // Phase 2: weighted sum of neighbor rows, 2 dims per lane
    const int dbase = 2 * lane;
    float f0 = 0.f, f1 = 0.f;
    #pragma unroll 8
    for (int k = 0; k < KNBR; ++k) {
        const float wk = (k < 32) ? __shfl(w0, k, 32) : __shfl(w1, k - 32, 32);
        const float* nr = h + (size_t)r[k] * DIM + head * HDIM + dbase;
        f0 += wk * nr[0];
        f1 += wk * nr[1];
    }
    float* o = fused + (size_t)s * DIM + head * HDIM + dbase;
    o[0] = f0;
    o[1] = f1;
}

// ---------------------------------------------------------------------------
extern "C" void kernel_launch(void* const* d_in, const int* in_sizes, int n_in,
                              void* d_out, int out_size, void* d_ws, size_t ws_size,
                              hipStream_t stream)
{
    const float* x      = (const float*)d_in[0];
    const int*   routes = (const int*)  d_in[1];
    const float* W_in   = (const float*)d_in[2];
    const float* W_out  = (const float*)d_in[3];
    const float* b_out  = (const float*)d_in[4];
    float*       out    = (float*)d_out;

    float* h     = (float*)d_ws;                    // S*D floats (8 MB)
    float* fused = h + (size_t)S_TOK * DIM;         // S*D floats (8 MB)

    const int waves  = (S_TOK / 32) * (DIM / 32);   // 2048 waves
    const int blocks = waves / 4;                   // 4 waves per block

    // h = x @ W_in^T
    gemm_nt_wmma<<<blocks, 128, 0, stream>>>(x, W_in, h, S_TOK, DIM, DIM,
                                             nullptr, nullptr);
    // gather attention -> fused
    attn_gather<<<S_TOK, 512, 0, stream>>>(h, routes, fused);
    // out = fused @ W_out^T + b_out + x
    gemm_nt_wmma<<<blocks, 128, 0, stream>>>(fused, W_out, out, S_TOK, DIM, DIM,
                                             b_out, x);
}